// HANMeta_1649267442137
// MI455X (gfx1250) — compile-verified
//
#include <hip/hip_runtime.h>
#include <stdint.h>

typedef unsigned int v4u __attribute__((ext_vector_type(4)));
typedef int          v4i __attribute__((ext_vector_type(4)));
typedef int          v8i __attribute__((ext_vector_type(8)));
typedef float        v4f __attribute__((ext_vector_type(4)));
typedef __attribute__((address_space(3))) v4f lds_v4f;

#define DIM   512   // D (== title emb dim)
#define KN    8     // K neighbors
#define PJOBS 64    // P

// wave32 xor-butterfly step via ds_swizzle (group-of-32 xor mask, §11.2):
// offset = (xor_mask << 5) | and_mask(0x1F)
template <int MASK>
__device__ static inline float swz_add(float v) {
  int x = __builtin_amdgcn_ds_swizzle(__float_as_int(v), MASK);
  return v + __int_as_float(x);
}

// Issue one TDM gather of 8 rows (each DIM f32 = 2KB) from a [mat_rows, DIM]
// f32 matrix at `base` into LDS at byte offset `lds_byte`.
// D# per CDNA5 ISA: gather_mode=1, 32-bit indices.
__device__ static inline void tdm_gather8(const float* base, unsigned mat_rows,
                                          unsigned lds_byte, const int idx[KN]) {
  unsigned long long ga = (unsigned long long)(uintptr_t)base;
  v4u g0;
  g0[0] = 0xC0000001u;                                   // gather_mode|idx32|count=1
  g0[1] = lds_byte;                                      // lds_addr (bytes)
  g0[2] = (unsigned)ga;                                  // global_addr[31:0]
  g0[3] = ((unsigned)(ga >> 32) & 0x01FFFFFFu)           // global_addr[56:32]
          | 0x80000000u;                                 // type=2 ("image")
  v8i g1;
  g1[0] = (int)(2u << 16);                               // wg_mask=0, data_size=4B
  g1[1] = (int)((unsigned)DIM << 16);                    // tensor_dim0[15:0]
  g1[2] = (int)((mat_rows & 0xFFFFu) << 16);             // dim0 hi=0 | tensor_dim1 lo
  g1[3] = (int)(((mat_rows >> 16) & 0xFFFFu)             // tensor_dim1 hi
                | ((unsigned)DIM << 16));                // tile_dim0 = 512
  g1[4] = KN;                                            // tile_dim1 = #valid indices
  g1[5] = DIM;                                           // tensor_dim0_stride lo32
  g1[6] = 0;                                             // stride hi | dim1_stride (ignored)
  g1[7] = 0;
  v4i g2 = { idx[0], idx[1], idx[2], idx[3] };           // row_index_0..3 (32-bit)
  v4i g3 = { idx[4], idx[5], idx[6], idx[7] };           // row_index_4..7
  v8i gx = { 0, 0, 0, 0, 0, 0, 0, 0 };                   // extra operand (clang-23 form)
  __builtin_amdgcn_tensor_load_to_lds(g0, g1, g2, g3, gx, 0);
}

__global__ __launch_bounds__(32) void han_meta_tdm(
    const float* __restrict__ inputs,      // [B*P, D]
    const float* __restrict__ titles,      // [T, D]
    const int*   __restrict__ nbr_batch,   // [B*P, K]
    const int*   __restrict__ nbr_job,
    const int*   __restrict__ nbr_title,
    const int*   __restrict__ nbr_mask,
    float* __restrict__ out,               // [B*P, 2*D]
    unsigned n_rows, unsigned n_titles)
{
  __shared__ __align__(16) float smem[2 * KN * DIM];     // 32KB: [nbr 8x512 | title 8x512]
  const unsigned r    = blockIdx.x;                      // one wave32 per (b,p) row
  const unsigned lane = threadIdx.x;

  // ---- wave-uniform neighbor metadata -> SGPRs --------------------------
  int nrow[KN], trow[KN], mk[KN];
  const unsigned ib = r * KN;
#pragma unroll
  for (int k = 0; k < KN; ++k) {
    int nb  = __builtin_amdgcn_readfirstlane(nbr_batch[ib + k]);
    int nj  = __builtin_amdgcn_readfirstlane(nbr_job[ib + k]);
    nrow[k] = nb * PJOBS + nj;                           // row into inputs[B*P, D]
    trow[k] = __builtin_amdgcn_readfirstlane(nbr_title[ib + k]);
    mk[k]   = __builtin_amdgcn_readfirstlane(nbr_mask[ib + k]);
  }

  // LDS byte offsets (generic-ptr low 32 bits == LDS address); launder so the
  // optimizer cannot prove smem is never written (the TDM writes it).
  unsigned off_n = (unsigned)(uintptr_t)&smem[0];
  unsigned off_t = (unsigned)(uintptr_t)&smem[KN * DIM];
  asm volatile("" : "+s"(off_n), "+s"(off_t));

  // ---- kick both DMA gathers, overlap with focal load -------------------
  tdm_gather8(inputs, n_rows,   off_n, nrow);            // TENSORcnt -> 1
  tdm_gather8(titles, n_titles, off_t, trow);            // TENSORcnt -> 2

  const float* frow = inputs + (size_t)r * DIM;          // focal: 16 f32/lane
  v4f focal[4];
#pragma unroll
  for (int j = 0; j < 4; ++j)
    focal[j] = *(const v4f*)(frow + lane * 4 + 128 * j); // coalesced b128 (RT: keep in L2)

  // ---- dots vs 8 neighbors (from LDS) -----------------------------------
  __builtin_amdgcn_s_wait_tensorcnt(1);                  // neighbor gather done (in-order)
  asm volatile("" ::: "memory");

  const lds_v4f* pn = (const lds_v4f*)off_n;
  float s[KN];
#pragma unroll
  for (int k = 0; k < KN; ++k) {
    v4f a = {0.f, 0.f, 0.f, 0.f};
#pragma unroll
    for (int j = 0; j < 4; ++j)
      a += focal[j] * pn[k * 128 + lane + 32 * j];       // ds_load_b128, conflict-free
    s[k] = a[0] + a[1] + a[2] + a[3];
  }
  // wave32 butterfly reduction via ds_swizzle -> all lanes hold all 8 dots
#pragma unroll
  for (int k = 0; k < KN; ++k) {
    float v = s[k];
    v = swz_add<0x21F>(v);   // SWAPX16
    v = swz_add<0x11F>(v);   // SWAPX8
    v = swz_add<0x09F>(v);   // SWAPX4
    v = swz_add<0x05F>(v);   // SWAPX2
    v = swz_add<0x03F>(v);   // SWAPX1
    s[k] = v;
  }

  // ---- masked softmax (all-masked row -> exact zeros, matches reference) -
  float mx = -1e30f;
#pragma unroll
  for (int k = 0; k < KN; ++k) if (mk[k]) mx = fmaxf(mx, s[k]);
  float e[KN]; float sum = 0.f;
#pragma unroll
  for (int k = 0; k < KN; ++k) { e[k] = mk[k] ? __expf(s[k] - mx) : 0.f; sum += e[k]; }
  float inv = (sum > 0.f) ? (1.0f / sum) : 0.f;

  // ---- weighted sum of title embeddings (from LDS) ----------------------
  __builtin_amdgcn_s_wait_tensorcnt(0);                  // title gather done
  asm volatile("" ::: "memory");

  const lds_v4f* pt = (const lds_v4f*)off_t;
  v4f acc[4] = { {0,0,0,0}, {0,0,0,0}, {0,0,0,0}, {0,0,0,0} };
#pragma unroll
  for (int k = 0; k < KN; ++k) {
    float w = e[k] * inv;
    v4f wv = {w, w, w, w};
#pragma unroll
    for (int j = 0; j < 4; ++j)
      acc[j] += pt[k * 128 + lane + 32 * j] * wv;
  }

  // ---- store [focal | graph_concat]: NT streaming stores so the 128MB
  // output does not evict the L2-resident inputs/title tables (164MB < 192MB)
  float* orow = out + (size_t)r * (2 * DIM);
#pragma unroll
  for (int j = 0; j < 4; ++j) {
    __builtin_nontemporal_store(focal[j], (v4f*)(orow + lane * 4 + 128 * j));
    __builtin_nontemporal_store(acc[j],   (v4f*)(orow + DIM + lane * 4 + 128 * j));
  }
}

extern "C" void kernel_launch(void* const* d_in, const int* in_sizes, int n_in,
                              void* d_out, int out_size, void* d_ws, size_t ws_size,
                              hipStream_t stream) {
  (void)n_in; (void)out_size; (void)d_ws; (void)ws_size;
  const float* inputs = (const float*)d_in[0];
  const float* titles = (const float*)d_in[1];
  const int*   nb     = (const int*)d_in[2];
  const int*   nj     = (const int*)d_in[3];
  const int*   nt     = (const int*)d_in[4];
  const int*   nm     = (const int*)d_in[5];
  float*       out    = (float*)d_out;

  unsigned n_rows   = (unsigned)(in_sizes[0] / DIM);   // B*P = 32768
  unsigned n_titles = (unsigned)(in_sizes[1] / DIM);   // T   = 50000

  hipLaunchKernelGGL(han_meta_tdm, dim3(n_rows), dim3(32), 0, stream,
                     inputs, titles, nb, nj, nt, nm, out, n_rows, n_titles);
}